// PrototypeKDEngine_78709570667268
// MI455X (gfx1250) — compile-verified
//
#include <hip/hip_runtime.h>
#include <hip/hip_bf16.h>
#include <math.h>

// ---------------- problem constants ----------------
#define NPIX   262144      // B*H*W = 8*128*256
#define HWSZ   32768       // H*W
#define CTOT   256         // channels
#define NCLS   19
#define IGN    255
#define NBLK3  2048        // blocks in main kernel (2048*8 waves = 16384 tiles = NPIX/16)

// output offsets (floats)
#define OUT_SIM   1
#define OUT_WL    (1 + NPIX)
#define OUT_MSIM  (1 + 2*NPIX)
#define OUT_PN    (2 + 2*NPIX)
#define OUT_PV    (2 + 2*NPIX + NCLS)

// workspace offsets (floats)
#define WS_SUMS   0           // 19*256
#define WS_CNT    4864        // 19 (pad to 4896)
#define WS_PROT   4896        // 19*256
#define WS_VALID  9760        // 19 (pad to 9792)
#define WS_PART   9792        // 4 * NBLK3

typedef float v2f __attribute__((ext_vector_type(2)));
typedef float v8f __attribute__((ext_vector_type(8)));

// ---------------- kernel 0: zero accumulators ----------------
__global__ void k_zero(float* __restrict__ ws, int n) {
  int i = blockIdx.x * 256 + threadIdx.x;
  if (i < n) ws[i] = 0.0f;
}

// ---------------- kernel 1: prototype accumulation ----------------
// 512 blocks x 256 threads, 2 pixels/thread. Lanes = consecutive pixels ->
// coalesced 128B loads per (channel, wave). Per-class partial sums live in
// LDS with stride 257 so bank = (cls + c) % 64: distinct classes in a wave
// hit distinct banks for ds_add_f32.
__global__ void k_proto_accum(const float* __restrict__ tf,
                              const int* __restrict__ mi,   // int64 mask, read low word
                              float* __restrict__ gsum, float* __restrict__ gcnt) {
  __shared__ float ldsS[NCLS * 257];
  __shared__ float ldsC[NCLS];
  const int tid = threadIdx.x;
  for (int i = tid; i < NCLS * 257; i += 256) ldsS[i] = 0.0f;
  if (tid < NCLS) ldsC[tid] = 0.0f;
  __syncthreads();

  for (int j = 0; j < 2; ++j) {
    int p  = (blockIdx.x * 2 + j) * 256 + tid;   // pixel id
    int mv = mi[2 * p];                          // low 32 bits of int64
    bool valid = (mv != IGN);
    int cls = valid ? mv : 0;
    if (valid) {
      atomicAdd(&ldsC[cls], 1.0f);
      int bb = p >> 15;                 // batch
      int hw = p & (HWSZ - 1);
      const float* sp = tf + (size_t)bb * (size_t)(CTOT * HWSZ) + hw;
      float* dst = &ldsS[cls * 257];
      for (int c = 0; c < CTOT; ++c) {
        atomicAdd(&dst[c], sp[(size_t)c * HWSZ]);
      }
    }
  }
  __syncthreads();
  for (int i = tid; i < NCLS * CTOT; i += 256) {
    int k = i >> 8, c = i & 255;
    atomicAdd(&gsum[i], ldsS[k * 257 + c]);
  }
  for (int k = tid; k < NCLS; k += 256) atomicAdd(&gcnt[k], ldsC[k]);
}

// ---------------- kernel 2: finalize prototypes ----------------
__global__ void k_proto_final(const float* __restrict__ gsum,
                              const float* __restrict__ gcnt,
                              float* __restrict__ protos,
                              float* __restrict__ pvalid,
                              float* __restrict__ out) {
  __shared__ float red[256];
  const int tid = threadIdx.x;   // channel
  for (int k = 0; k < NCLS; ++k) {
    float cnt  = gcnt[k];
    float s    = gsum[k * CTOT + tid];
    float mean = (cnt > 0.0f) ? (s / fmaxf(cnt, 1.0f)) : 0.0f;
    red[tid] = mean * mean;
    __syncthreads();
    for (int st = 128; st > 0; st >>= 1) {
      if (tid < st) red[tid] += red[tid + st];
      __syncthreads();
    }
    float r = sqrtf(red[0]);
    __syncthreads();                       // before red is rewritten next iter
    float q = mean / fmaxf(r, 1e-6f);      // normalized prototype
    protos[k * CTOT + tid] = q;
    if (tid == 0) {
      float pn = r / fmaxf(r, 1e-6f);      // ||normalized proto||
      out[OUT_PN + k] = pn;
      float v = (pn > 0.0f) ? 1.0f : 0.0f;
      out[OUT_PV + k] = v;
      pvalid[k] = v;
    }
  }
}

// ---------------- kernel 3: WMMA similarity + loss maps ----------------
// Each wave computes a 16-pixel x 19-class similarity GEMM via
// V_WMMA_F32_16X16X4_F32 (K-loop over 256 channels, two class halves).
// B operands live in LDS pre-swizzled as (k,k+1) channel pairs per class so
// each operand is a single 8B-aligned ds_load_b64 straight into an even VGPR
// pair (no marshaling movs). Per-pixel ||s||^2 is accumulated on the same
// global loads and folded across lane halves with __shfl_xor(...,16).
__global__ void __launch_bounds__(256)
k_main(const float* __restrict__ sf, const int* __restrict__ mi,
       const float* __restrict__ swt,
       const float* __restrict__ protos, const float* __restrict__ pvalid,
       float* __restrict__ out, float* __restrict__ part) {
  // ldsB[kp*66 + 2*class + (k&1)] = proto[class][k], kp = k/2, class padded to 32
  __shared__ __align__(16) float ldsB[128 * 66];
  __shared__ float ldsT[8 * 512];     // per-wave 16x32 D spill
  __shared__ float ldsV[NCLS];
  __shared__ float red[4][256];

  const int tid = threadIdx.x;
  for (int i = tid; i < CTOT * 32; i += 256) {
    int k = i >> 5, n = i & 31;
    ldsB[(k >> 1) * 66 + 2 * n + (k & 1)] = (n < NCLS) ? protos[n * CTOT + k] : 0.0f;
  }
  if (tid < NCLS) ldsV[tid] = pvalid[tid];
  __syncthreads();

  const int wave = tid >> 5, lane = tid & 31;
  const int h = lane >> 4, ln = lane & 15;
  const int tile = blockIdx.x * 8 + wave;
  const int m = tile * 16;            // first pixel of tile
  const int p0 = m + ln;              // this lane's pixel row (A matrix M=ln)
  const int bb = p0 >> 15;
  const int hw = p0 & (HWSZ - 1);
  const float* sp = sf + (size_t)bb * (size_t)(CTOT * HWSZ) + hw
                       + (size_t)(2 * h) * HWSZ;   // lane half covers K%4 = 2h,2h+1
  // B pointer: kp = k0/2 + h, advances by 2 rows (132 floats) per k-step
  const float* bp = &ldsB[h * 66 + 2 * ln];

  v8f d1 = {};   // classes 0..15
  v8f d2 = {};   // classes 16..31 (19..31 are zero-padded protos)
  float acc = 0.0f;
  #pragma unroll 4
  for (int k0 = 0; k0 < CTOT; k0 += 4) {
    v2f a;
    a.x = sp[0];
    a.y = sp[HWSZ];
    sp += (size_t)4 * HWSZ;
    v2f b1 = *(const v2f*)(bp);        // classes  0..15, channels (k0+2h, k0+2h+1)
    v2f b2 = *(const v2f*)(bp + 32);   // classes 16..31, same channels
    bp += 132;
    acc += a.x * a.x + a.y * a.y;
    d1 = __builtin_amdgcn_wmma_f32_16x16x4_f32(false, a, false, b1, (short)0, d1, false, false);
    d2 = __builtin_amdgcn_wmma_f32_16x16x4_f32(false, a, false, b2, (short)0, d2, false, false);
  }
  // fold ||s||^2 halves (K%4 in {0,1} lives in lanes 0-15, {2,3} in 16-31)
  float tot = acc + __shfl_xor(acc, 16);

  // spill D: VGPR r holds M=r (lanes 0-15) / M=r+8 (lanes 16-31), N=lane&15
  float* T = &ldsT[wave * 512];
  #pragma unroll
  for (int r = 0; r < 8; ++r) {
    int row = r + 8 * h;
    T[row * 32 + ln]      = d1[r];
    T[row * 32 + 16 + ln] = d2[r];
  }

  float q0 = 0.0f, q1 = 0.0f, q2 = 0.0f, q3 = 0.0f;
  if (lane < 16) {
    int p   = m + lane;
    int mv  = mi[2 * p];
    int cls = (mv == IGN) ? 0 : mv;
    float dot = T[lane * 32 + cls];
    float sim = dot / fmaxf(sqrtf(tot), 1e-12f);
    float vf  = ((mv != IGN) && (ldsV[cls] > 0.0f)) ? 1.0f : 0.0f;
    float w   = swt[p];
    float wl  = (1.0f - sim) * vf * w;
    out[OUT_SIM + p] = sim;
    out[OUT_WL + p]  = wl;
    q0 = wl; q1 = vf * w; q2 = vf; q3 = sim * vf;
  }
  red[0][tid] = q0; red[1][tid] = q1; red[2][tid] = q2; red[3][tid] = q3;
  __syncthreads();
  for (int st = 128; st > 0; st >>= 1) {
    if (tid < st) {
      red[0][tid] += red[0][tid + st];
      red[1][tid] += red[1][tid + st];
      red[2][tid] += red[2][tid + st];
      red[3][tid] += red[3][tid + st];
    }
    __syncthreads();
  }
  if (tid == 0) {
    part[0 * NBLK3 + blockIdx.x] = red[0][0];
    part[1 * NBLK3 + blockIdx.x] = red[1][0];
    part[2 * NBLK3 + blockIdx.x] = red[2][0];
    part[3 * NBLK3 + blockIdx.x] = red[3][0];
  }
}

// ---------------- kernel 4: final scalars ----------------
__global__ void k_final(const float* __restrict__ part, float* __restrict__ out) {
  __shared__ float red[4][256];
  const int tid = threadIdx.x;
  float s0 = 0, s1 = 0, s2 = 0, s3 = 0;
  for (int i = tid; i < NBLK3; i += 256) {
    s0 += part[0 * NBLK3 + i];
    s1 += part[1 * NBLK3 + i];
    s2 += part[2 * NBLK3 + i];
    s3 += part[3 * NBLK3 + i];
  }
  red[0][tid] = s0; red[1][tid] = s1; red[2][tid] = s2; red[3][tid] = s3;
  __syncthreads();
  for (int st = 128; st > 0; st >>= 1) {
    if (tid < st) {
      red[0][tid] += red[0][tid + st];
      red[1][tid] += red[1][tid + st];
      red[2][tid] += red[2][tid + st];
      red[3][tid] += red[3][tid + st];
    }
    __syncthreads();
  }
  if (tid == 0) {
    out[0]        = red[0][0] / fmaxf(red[1][0], 1.0f);   // final_loss
    out[OUT_MSIM] = red[3][0] / fmaxf(red[2][0], 1.0f);   // mean_sim
  }
}

// ---------------- launch ----------------
extern "C" void kernel_launch(void* const* d_in, const int* in_sizes, int n_in,
                              void* d_out, int out_size, void* d_ws, size_t ws_size,
                              hipStream_t stream) {
  const float* sf  = (const float*)d_in[0];   // s_feat
  const float* tf  = (const float*)d_in[1];   // t_feat
  const int*   mi  = (const int*)d_in[2];     // mask (int64, read low words)
  const float* swt = (const float*)d_in[3];   // swt_weight

  float* out = (float*)d_out;
  float* ws  = (float*)d_ws;
  float* gsum   = ws + WS_SUMS;
  float* gcnt   = ws + WS_CNT;
  float* protos = ws + WS_PROT;
  float* pvalid = ws + WS_VALID;
  float* part   = ws + WS_PART;

  k_zero<<<(WS_PROT + 255) / 256, 256, 0, stream>>>(ws, WS_PROT); // sums + counts
  k_proto_accum<<<512, 256, 0, stream>>>(tf, mi, gsum, gcnt);
  k_proto_final<<<1, 256, 0, stream>>>(gsum, gcnt, protos, pvalid, out);
  k_main<<<NBLK3, 256, 0, stream>>>(sf, mi, swt, protos, pvalid, out, part);
  k_final<<<1, 256, 0, stream>>>(part, out);
}